// POSE_23210003267897
// MI455X (gfx1250) — compile-verified
//
#include <hip/hip_runtime.h>
#include <math.h>

typedef float v2f __attribute__((ext_vector_type(2)));
typedef float v8f __attribute__((ext_vector_type(8)));

#define W_DIM     1920
#define FX_C      1000.0f
#define FY_C      1000.0f
#define CX_C      960.0f
#define CY_C      540.0f
#define BOUND_C   2.0f
#define GRID_N    128
#define MAX_STEPS 128
#define MIN_NEAR  0.05f

// ---------------------------------------------------------------------------
// Kernel 1: pose composition. One wave32, EXEC all ones (WMMA requirement).
// Computes Rr,tr = exp(se3[index]); then Rn = Rb @ Rr and Rb @ tr with one
// V_WMMA_F32_16X16X4_F32:  A(16x4) = Rb padded, B(4x16) = [Rr | tr] padded.
// f32 A layout: lane l<16 holds A[l][0..1] in v0,v1; lane l+16 holds A[l][2..3].
// f32 B layout: v0 = B[K=0|2][lane&15], v1 = B[K=1|3][lane&15] (K+2 for upper lanes).
// D layout: lanes 0-15, VGPR r = D[M=r][N=lane].
// ---------------------------------------------------------------------------
__global__ __launch_bounds__(32) void pose_compose_kernel(
    const float* __restrict__ se3, const float* __restrict__ poses,
    const int* __restrict__ index_p, float* __restrict__ pose_out)
{
    const int lane = threadIdx.x;
    const int idx  = index_p[0];
    const float* s = se3 + idx * 6;
    const float w0 = s[0], w1 = s[1], w2 = s[2];
    const float u0 = s[3], u1 = s[4], u2 = s[5];

    float th = fmaxf(sqrtf(w0*w0 + w1*w1 + w2*w2), 1e-8f);
    float sth = sinf(th), cth = cosf(th);
    float Af = sth / th;
    float Bf = (1.0f - cth) / (th * th);
    float Cf = (th - sth) / (th * th * th);

    float Wm[3][3] = {{0.f, -w2,  w1},
                      { w2, 0.f, -w0},
                      {-w1,  w0, 0.f}};
    float W2[3][3];
    for (int i = 0; i < 3; ++i)
        for (int j = 0; j < 3; ++j) {
            float acc = 0.f;
            for (int k = 0; k < 3; ++k) acc += Wm[i][k] * Wm[k][j];
            W2[i][j] = acc;
        }
    float Rr[3][3], Vm[3][3];
    for (int i = 0; i < 3; ++i)
        for (int j = 0; j < 3; ++j) {
            float I = (i == j) ? 1.f : 0.f;
            Rr[i][j] = I + Af * Wm[i][j] + Bf * W2[i][j];
            Vm[i][j] = I + Bf * Wm[i][j] + Cf * W2[i][j];
        }
    const float uu[3] = {u0, u1, u2};
    float tr[3];
    for (int i = 0; i < 3; ++i)
        tr[i] = Vm[i][0]*uu[0] + Vm[i][1]*uu[1] + Vm[i][2]*uu[2];

    const float* bp = poses + idx * 12;   // poses[p] is 3x4 row-major: R | t

    // Build A = Rb (zero-padded 16x4)
    const int row = lane & 15;
    const int kb  = (lane >> 4) * 2;      // 0 or 2
    v2f a;
    a.x = (row < 3 && kb     < 3) ? bp[row*4 + kb    ] : 0.f;
    a.y = (row < 3 && kb + 1 < 3) ? bp[row*4 + kb + 1] : 0.f;

    // Build B = [Rr | tr] (zero-padded 4x16); column 3 carries tr
    const int col = lane & 15;
    float b0 = 0.f, b1 = 0.f;
    if (kb < 3)     b0 = (col < 3) ? Rr[kb][col]     : ((col == 3) ? tr[kb]     : 0.f);
    if (kb + 1 < 3) b1 = (col < 3) ? Rr[kb + 1][col] : ((col == 3) ? tr[kb + 1] : 0.f);
    v2f b; b.x = b0; b.y = b1;

    v8f c = {};
    v8f d = __builtin_amdgcn_wmma_f32_16x16x4_f32(
        false, a, false, b, (short)0, c, false, false);

    if (lane < 3) {               // columns 0..2 of Rn, rows 0..2 in d[0..2]
        pose_out[0*3 + lane] = d[0];
        pose_out[1*3 + lane] = d[1];
        pose_out[2*3 + lane] = d[2];
    } else if (lane == 3) {       // column 3 = Rb @ tr; add tb
        pose_out[9 + 0] = d[0] + bp[0*4 + 3];
        pose_out[9 + 1] = d[1] + bp[1*4 + 3];
        pose_out[9 + 2] = d[2] + bp[2*4 + 3];
    }
}

// ---------------------------------------------------------------------------
// Kernel 2: one wave32 per ray; lane handles steps {lane, lane+32, +64, +96}.
// Writes xyzs_m / dirs_m / deltas / near / far and per-ray count (int bits
// into rays_info[...,1]).
// ---------------------------------------------------------------------------
__global__ __launch_bounds__(256) void ray_march_kernel(
    const float* __restrict__ grid, const int* __restrict__ pix_inds,
    const float* __restrict__ pose,
    float* __restrict__ out_xyz, float* __restrict__ out_dir,
    float* __restrict__ out_del, int* __restrict__ out_info,
    float* __restrict__ out_near, float* __restrict__ out_far,
    int n_rays)
{
    const int wave = (int)((blockIdx.x * blockDim.x + threadIdx.x) >> 5);
    const int lane = threadIdx.x & 31;
    if (wave >= n_rays) return;

    // uniform pose -> scalar loads
    const float R00 = pose[0], R01 = pose[1], R02 = pose[2];
    const float R10 = pose[3], R11 = pose[4], R12 = pose[5];
    const float R20 = pose[6], R21 = pose[7], R22 = pose[8];
    const float ox  = pose[9], oy  = pose[10], oz = pose[11];

    const int pix = pix_inds[wave];
    float fi = (float)(pix % W_DIM) + 0.5f;
    float fj = (float)(pix / W_DIM) + 0.5f;
    float dcx = (fi - CX_C) / FX_C;
    float dcy = (fj - CY_C) / FY_C;
    float inv_n = 1.0f / sqrtf(dcx*dcx + dcy*dcy + 1.0f);
    float dcz = inv_n;
    dcx *= inv_n; dcy *= inv_n;

    const float dx = R00*dcx + R01*dcy + R02*dcz;
    const float dy = R10*dcx + R11*dcy + R12*dcz;
    const float dz = R20*dcx + R21*dcy + R22*dcz;

    // AABB slab test (replicates where(|d|>1e-9, d, 1e-9))
    float lo[3], hi[3];
    const float dv[3] = {dx, dy, dz};
    const float ov[3] = {ox, oy, oz};
    for (int ax = 0; ax < 3; ++ax) {
        float dd  = (fabsf(dv[ax]) > 1e-9f) ? dv[ax] : 1e-9f;
        float inv = 1.0f / dd;
        float t1  = (-BOUND_C - ov[ax]) * inv;
        float t2  = ( BOUND_C - ov[ax]) * inv;
        lo[ax] = fminf(t1, t2);
        hi[ax] = fmaxf(t1, t2);
    }
    float near = fmaxf(fmaxf(fmaxf(lo[0], lo[1]), lo[2]), MIN_NEAR);
    float far  = fmaxf(fminf(fminf(hi[0], hi[1]), hi[2]), near + 0.0001f);
    float dt   = (far - near) / (float)MAX_STEPS;

    if (lane == 0) { out_near[wave] = near; out_far[wave] = far; }

    const int base3 = wave * (MAX_STEPS * 3);
    const int base2 = wave * (MAX_STEPS * 2);
    int cnt = 0;

#pragma unroll
    for (int it = 0; it < MAX_STEPS / 32; ++it) {
        const int step = it * 32 + lane;
        const float ts = near + ((float)step + 0.5f) * dt;
        const float x = ox + dx * ts;
        const float y = oy + dy * ts;
        const float z = oz + dz * ts;
        int ix = (int)((x + BOUND_C) * 32.0f);
        int iy = (int)((y + BOUND_C) * 32.0f);
        int iz = (int)((z + BOUND_C) * 32.0f);
        ix = min(max(ix, 0), GRID_N - 1);
        iy = min(max(iy, 0), GRID_N - 1);
        iz = min(max(iz, 0), GRID_N - 1);
        const float g = grid[(ix * GRID_N + iy) * GRID_N + iz];
        const bool occ = g > 0.5f;
        const float m = occ ? 1.0f : 0.0f;

        const int o3 = base3 + step * 3;
        out_xyz[o3 + 0] = x * m;
        out_xyz[o3 + 1] = y * m;
        out_xyz[o3 + 2] = z * m;
        out_dir[o3 + 0] = dx * m;
        out_dir[o3 + 1] = dy * m;
        out_dir[o3 + 2] = dz * m;
        const int o2 = base2 + step * 2;
        out_del[o2 + 0] = dt * m;
        out_del[o2 + 1] = ts * m;
        cnt += occ ? 1 : 0;
    }

    // wave32 reduction of hit count
    for (int offs = 16; offs > 0; offs >>= 1)
        cnt += __shfl_xor(cnt, offs, 32);
    if (lane == 0) out_info[wave * 2 + 1] = cnt;
}

// ---------------------------------------------------------------------------
// Kernel 3: exclusive prefix sum over counts -> rays_info[...,0].
// Single 1024-thread block; 64 rays per thread + LDS Hillis-Steele scan.
// ---------------------------------------------------------------------------
__global__ __launch_bounds__(1024) void scan_kernel(int* __restrict__ info, int n_rays)
{
    __shared__ int sdata[1024];
    const int tid = threadIdx.x;
    const int per = (n_rays + 1023) / 1024;
    const int begin = tid * per;
    const int end = min(begin + per, n_rays);

    int sum = 0;
    for (int i = begin; i < end; ++i) sum += info[i * 2 + 1];
    sdata[tid] = sum;
    __syncthreads();

    for (int off = 1; off < 1024; off <<= 1) {
        int v = (tid >= off) ? sdata[tid - off] : 0;
        __syncthreads();
        sdata[tid] += v;
        __syncthreads();
    }
    int run = sdata[tid] - sum;   // exclusive prefix for this chunk
    for (int i = begin; i < end; ++i) {
        int c = info[i * 2 + 1];
        info[i * 2 + 0] = run;
        run += c;
    }
}

// ---------------------------------------------------------------------------
extern "C" void kernel_launch(void* const* d_in, const int* in_sizes, int n_in,
                              void* d_out, int out_size, void* d_ws, size_t ws_size,
                              hipStream_t stream) {
    const float* se3     = (const float*)d_in[0];
    const float* poses   = (const float*)d_in[1];
    const float* grid    = (const float*)d_in[2];
    const int*   pix     = (const int*)d_in[3];
    const int*   index_p = (const int*)d_in[4];

    float* out = (float*)d_out;
    const int n_rays = in_sizes[3];

    // Output layout (tuple concatenation order)
    const size_t n3       = (size_t)n_rays * MAX_STEPS * 3;  // xyzs_m
    const size_t n2       = (size_t)n_rays * MAX_STEPS * 2;  // deltas
    const size_t off_dir  = n3;
    const size_t off_del  = 2 * n3;
    const size_t off_info = off_del + n2;
    const size_t off_near = off_info + (size_t)n_rays * 2;
    const size_t off_far  = off_near + (size_t)n_rays;

    float* pose_ws = (float*)d_ws;   // 12 floats

    pose_compose_kernel<<<1, 32, 0, stream>>>(se3, poses, index_p, pose_ws);

    const int waves_per_block = 8;   // 256 threads = 8 wave32s
    const int blocks = (n_rays + waves_per_block - 1) / waves_per_block;
    ray_march_kernel<<<blocks, 256, 0, stream>>>(
        grid, pix, pose_ws,
        out, out + off_dir, out + off_del,
        (int*)(out + off_info), out + off_near, out + off_far, n_rays);

    scan_kernel<<<1, 1024, 0, stream>>>((int*)(out + off_info), n_rays);
}